// GNN_model_11063835755072
// MI455X (gfx1250) — compile-verified
//
#include <hip/hip_runtime.h>

typedef float v2f __attribute__((ext_vector_type(2)));
typedef float v8f __attribute__((ext_vector_type(8)));

#define HDIM 128  // hidden width (H) — fixed by the model

// ---------------- normalization ----------------

__global__ void k_init_deg(float* __restrict__ deg, int n) {
    int i = blockIdx.x * blockDim.x + threadIdx.x;
    if (i < n) deg[i] = 1.0f;  // self-loop contributes 1 to every node's degree
}

__global__ void k_deg_scatter(const int* __restrict__ dst, float* __restrict__ deg, int e) {
    int i = blockIdx.x * blockDim.x + threadIdx.x;
    if (i < e) atomicAdd(&deg[dst[i]], 1.0f);
}

__global__ void k_dinv(float* __restrict__ deg, int n) {
    int i = blockIdx.x * blockDim.x + threadIdx.x;
    if (i < n) {
        float d = deg[i];
        deg[i] = (d > 0.0f) ? rsqrtf(d) : 0.0f;  // in-place: deg -> dinv
    }
}

__global__ void k_norm(const int* __restrict__ src, const int* __restrict__ dst,
                       const float* __restrict__ dinv, float* __restrict__ norm, int e) {
    int i = blockIdx.x * blockDim.x + threadIdx.x;
    if (i < e) norm[i] = dinv[src[i]] * dinv[dst[i]];
}

// ---------------- dense GEMM via V_WMMA_F32_16X16X4_F32 ----------------
// C[M x 128] = (reluA ? relu(A) : A)[M x K] @ W[K x 128]
// One block = one 16-row tile; 8 waves cover the 8 column tiles of H=128.
// A-tile staged in LDS with padded stride (K+1) for conflict-free reads.
__global__ __launch_bounds__(256) void k_gemm_wmma(const float* __restrict__ A,
                                                   const float* __restrict__ W,
                                                   float* __restrict__ C,
                                                   int K, int kshift, int reluA) {
    __shared__ float As[16 * 129];  // max: K=128 -> stride 129
    const int stride = K + 1;
    const int row0 = blockIdx.x << 4;
    const int tid = threadIdx.x;

    // cooperative stage of the 16 x K A-tile (optionally fused ReLU)
    const int total = 16 << kshift;
    const int cmask = K - 1;
    for (int i = tid; i < total; i += 256) {
        int r = i >> kshift;
        int c = i & cmask;
        float v = A[((size_t)(row0 + r) << kshift) + c];
        if (reluA) v = fmaxf(v, 0.0f);
        As[r * stride + c] = v;
    }
    __syncthreads();

    const int lane = tid & 31;
    const int col0 = (tid >> 5) << 4;      // wave id * 16
    const int m    = lane & 15;            // row (A) / col (B,D) within tile
    const int half = lane >> 4;            // 0 or 1
    const int kb   = half << 1;            // K-offset within a 4-step: 0 or 2

    v8f acc = {};
    for (int k = 0; k < K; k += 4) {
        // A fragment (16x4 f32): lanes 0-15 -> K = k+0,k+1 ; lanes 16-31 -> K = k+2,k+3
        v2f a;
        a.x = As[m * stride + k + kb];
        a.y = As[m * stride + k + kb + 1];
        // B fragment (4x16 f32): same K split, N = lane&15
        v2f b;
        b.x = W[(k + kb) * HDIM + col0 + m];
        b.y = W[(k + kb + 1) * HDIM + col0 + m];
        // 8 args: (neg_a, A, neg_b, B, c_mod, C, reuse_a, reuse_b)
        acc = __builtin_amdgcn_wmma_f32_16x16x4_f32(false, a, false, b,
                                                    (short)0, acc, false, false);
    }

    // D layout: VGPR v, lanes 0-15 -> M = v ; lanes 16-31 -> M = v+8 ; N = lane&15
#pragma unroll
    for (int r = 0; r < 8; ++r) {
        C[(size_t)(row0 + r + 8 * half) * HDIM + col0 + m] = acc[r];
    }
}

// ---------------- aggregation ----------------

// agg[n,f] = dinv[n]^2 * h[n,f] + bias[f]   (self-loop term + bias; full overwrite)
__global__ void k_self_bias(const float* __restrict__ h, const float* __restrict__ dinv,
                            const float* __restrict__ bias, float* __restrict__ agg, int n) {
    int i = blockIdx.x * blockDim.x + threadIdx.x;  // over N*H
    if (i < n) {
        int node = i >> 7;
        int f = i & 127;
        float di = dinv[node];
        agg[i] = di * di * h[i] + bias[f];
    }
}

// one wave32 per edge; each lane moves a float4 (H=128 = 32 lanes * 4)
__global__ void k_edge_scatter(const int* __restrict__ src, const int* __restrict__ dst,
                               const float* __restrict__ norm, const float* __restrict__ h,
                               float* __restrict__ agg, int e) {
    int t = blockIdx.x * blockDim.x + threadIdx.x;
    int ei = t >> 5;
    int lane = t & 31;
    if (ei >= e) return;
    int s = src[ei], d = dst[ei];
    float w = norm[ei];
    float4 hv = ((const float4*)(h + ((size_t)s << 7)))[lane];
    float* ap = agg + ((size_t)d << 7) + (lane << 2);
    atomicAdd(ap + 0, w * hv.x);
    atomicAdd(ap + 1, w * hv.y);
    atomicAdd(ap + 2, w * hv.z);
    atomicAdd(ap + 3, w * hv.w);
}

// ---------------- pooling + readout ----------------

__global__ void k_zero(float* __restrict__ p, int n) {
    int i = blockIdx.x * blockDim.x + threadIdx.x;
    if (i < n) p[i] = 0.0f;
}

__global__ void k_pool_scatter(const int* __restrict__ batch, const float* __restrict__ h,
                               float* __restrict__ pooled, float* __restrict__ cnt, int n) {
    int t = blockIdx.x * blockDim.x + threadIdx.x;
    int node = t >> 5;
    int lane = t & 31;
    if (node >= n) return;
    int g = batch[node];
    float4 hv = ((const float4*)(h + ((size_t)node << 7)))[lane];
    float* pp = pooled + ((size_t)g << 7) + (lane << 2);
    atomicAdd(pp + 0, hv.x);
    atomicAdd(pp + 1, hv.y);
    atomicAdd(pp + 2, hv.z);
    atomicAdd(pp + 3, hv.w);
    if (lane == 0) atomicAdd(&cnt[g], 1.0f);
}

// out[g] = (pooled[g,:] . Wl) / max(cnt[g],1) + bl   — one wave per graph
__global__ void k_readout(const float* __restrict__ pooled, const float* __restrict__ cnt,
                          const float* __restrict__ Wl, const float* __restrict__ bl,
                          float* __restrict__ out) {
    int g = blockIdx.x;
    int lane = threadIdx.x;
    float4 pv = ((const float4*)(pooled + ((size_t)g << 7)))[lane];
    float4 wv = ((const float4*)Wl)[lane];
    float s = pv.x * wv.x + pv.y * wv.y + pv.z * wv.z + pv.w * wv.w;
#pragma unroll
    for (int off = 16; off > 0; off >>= 1) s += __shfl_xor(s, off, 32);
    if (lane == 0) out[g] = s / fmaxf(cnt[g], 1.0f) + bl[0];
}

// ---------------- launch ----------------

extern "C" void kernel_launch(void* const* d_in, const int* in_sizes, int n_in,
                              void* d_out, int out_size, void* d_ws, size_t ws_size,
                              hipStream_t stream) {
    const float* x     = (const float*)d_in[0];
    const int*   ei    = (const int*)d_in[1];
    const int*   batch = (const int*)d_in[2];
    const float* W1 = (const float*)d_in[3];
    const float* b1 = (const float*)d_in[4];
    const float* W2 = (const float*)d_in[5];
    const float* b2 = (const float*)d_in[6];
    const float* W3 = (const float*)d_in[7];
    const float* b3 = (const float*)d_in[8];
    const float* Wl = (const float*)d_in[9];
    const float* bl = (const float*)d_in[10];
    float* out = (float*)d_out;

    const int N = in_sizes[2];
    const int E = in_sizes[1] / 2;
    const int F = in_sizes[0] / N;   // 64
    const int H = HDIM;              // 128
    const int G = out_size;          // 512

    const int* src = ei;
    const int* dst = ei + E;

    // workspace carve (256B-aligned)
    char* wp = (char*)d_ws;
    auto carve = [&](size_t bytes) -> float* {
        float* p = (float*)wp;
        wp += (bytes + 255) & ~(size_t)255;
        return p;
    };
    float* dinv   = carve((size_t)N * 4);
    float* norm   = carve((size_t)E * 4);
    float* buf0   = carve((size_t)N * H * 4);  // GEMM output (pre-aggregation h)
    float* buf1   = carve((size_t)N * H * 4);  // aggregated output
    float* pooled = carve((size_t)G * H * 4);
    float* cnt    = carve((size_t)G * 4);

    auto gridFor = [](long long n, int b) -> int { return (int)((n + b - 1) / b); };
    const int kF = (F == 64) ? 6 : 7;
    const int kH = 7;

    // --- normalization: deg -> dinv -> edge norms ---
    k_init_deg   <<<gridFor(N, 256), 256, 0, stream>>>(dinv, N);
    k_deg_scatter<<<gridFor(E, 256), 256, 0, stream>>>(dst, dinv, E);
    k_dinv       <<<gridFor(N, 256), 256, 0, stream>>>(dinv, N);
    k_norm       <<<gridFor(E, 256), 256, 0, stream>>>(src, dst, dinv, norm, E);

    // --- layer 1 ---
    k_gemm_wmma   <<<N / 16, 256, 0, stream>>>(x, W1, buf0, F, kF, 0);
    k_self_bias   <<<gridFor((long long)N * H, 256), 256, 0, stream>>>(buf0, dinv, b1, buf1, N * H);
    k_edge_scatter<<<gridFor((long long)E * 32, 256), 256, 0, stream>>>(src, dst, norm, buf0, buf1, E);

    // --- layer 2 (ReLU of layer-1 output fused into GEMM A-read) ---
    k_gemm_wmma   <<<N / 16, 256, 0, stream>>>(buf1, W2, buf0, H, kH, 1);
    k_self_bias   <<<gridFor((long long)N * H, 256), 256, 0, stream>>>(buf0, dinv, b2, buf1, N * H);
    k_edge_scatter<<<gridFor((long long)E * 32, 256), 256, 0, stream>>>(src, dst, norm, buf0, buf1, E);

    // --- layer 3 (no ReLU after) ---
    k_gemm_wmma   <<<N / 16, 256, 0, stream>>>(buf1, W3, buf0, H, kH, 1);
    k_self_bias   <<<gridFor((long long)N * H, 256), 256, 0, stream>>>(buf0, dinv, b3, buf1, N * H);
    k_edge_scatter<<<gridFor((long long)E * 32, 256), 256, 0, stream>>>(src, dst, norm, buf0, buf1, E);

    // --- global mean pool + linear readout ---
    k_zero        <<<gridFor((long long)G * H, 256), 256, 0, stream>>>(pooled, G * H);
    k_zero        <<<gridFor(G, 256), 256, 0, stream>>>(cnt, G);
    k_pool_scatter<<<gridFor((long long)N * 32, 256), 256, 0, stream>>>(batch, buf1, pooled, cnt, N);
    k_readout     <<<G, 32, 0, stream>>>(pooled, cnt, Wl, bl, out);
}